// GCNConv_38543036514348
// MI455X (gfx1250) — compile-verified
//
#include <hip/hip_runtime.h>
#include <hip/hip_bf16.h>

typedef __attribute__((ext_vector_type(2))) float v2f;
typedef __attribute__((ext_vector_type(8))) float v8f;

#define IN_DIM  128
#define OUT_DIM 64
#define LDS_STRIDE 72   // 64 + 8 pad: 2-row offset = 144 floats -> bank shift 16
                        // -> half-waves use disjoint 16-bank groups (no conflicts)

// ---------------------------------------------------------------------------
// Kernel 1: H = X @ W via V_WMMA_F32_16X16X4_F32 (full f32 precision).
// W (32 KB) is staged once per block into LDS (padded stride 72 floats);
// each wave then computes a 16-row x 64-col strip with 4 v8f accumulators,
// reading A from global (streaming X) and B fragments from LDS.
// A-fragment (16x4 f32): lane m=lane&15; lanes<16 carry K=k0,k0+1; lanes>=16
// carry K=k0+2,k0+3 (v[0]=even K, v[1]=odd K).
// B-fragment (4x16 f32): same K split across lane halves, N = lane&15.
// C/D (16x16 f32): VGPR i -> row i (lanes 0-15) / row i+8 (lanes 16-31).
// ---------------------------------------------------------------------------
__global__ void gcn_gemm_wmma(const float* __restrict__ X,
                              const float* __restrict__ W,
                              float* __restrict__ H,
                              int tilesM) {
    __shared__ float WL[IN_DIM * LDS_STRIDE];   // 36,864 B of 320 KB WGP LDS

    // cooperative coalesced staging of W[128][64] -> WL[128][72]
    for (int idx = threadIdx.x; idx < IN_DIM * OUT_DIM; idx += blockDim.x) {
        const int k = idx >> 6;          // / OUT_DIM
        const int n = idx & (OUT_DIM - 1);
        WL[k * LDS_STRIDE + n] = W[idx];
    }
    __syncthreads();

    const int lane = threadIdx.x & 31;
    const int wave = (blockIdx.x * (blockDim.x >> 5)) + (threadIdx.x >> 5);
    if (wave >= tilesM) return;           // wave-uniform: EXEC stays all-1s

    const int m0    = wave * 16;
    const int mrow  = m0 + (lane & 15);   // A row this lane feeds
    const int khalf = (lane >> 4) * 2;    // 0 for lanes 0-15, 2 for lanes 16-31
    const int ncol  = lane & 15;          // B column this lane feeds

    v8f acc0 = {}, acc1 = {}, acc2 = {}, acc3 = {};

    const float* __restrict__ xrow = X + (size_t)mrow * IN_DIM;

    #pragma unroll 4
    for (int k0 = 0; k0 < IN_DIM; k0 += 4) {
        // A: two consecutive f32 at even offset -> aligned 8B global load
        v2f a = *(const v2f*)(xrow + k0 + khalf);

        const float* wl0 = WL + (k0 + khalf) * LDS_STRIDE;   // K = k0+khalf
        const float* wl1 = wl0 + LDS_STRIDE;                 // K = k0+khalf+1

        v2f b0 = { wl0[ncol +  0], wl1[ncol +  0] };
        v2f b1 = { wl0[ncol + 16], wl1[ncol + 16] };
        v2f b2 = { wl0[ncol + 32], wl1[ncol + 32] };
        v2f b3 = { wl0[ncol + 48], wl1[ncol + 48] };

        acc0 = __builtin_amdgcn_wmma_f32_16x16x4_f32(false, a, false, b0, (short)0, acc0, false, false);
        acc1 = __builtin_amdgcn_wmma_f32_16x16x4_f32(false, a, false, b1, (short)0, acc1, false, false);
        acc2 = __builtin_amdgcn_wmma_f32_16x16x4_f32(false, a, false, b2, (short)0, acc2, false, false);
        acc3 = __builtin_amdgcn_wmma_f32_16x16x4_f32(false, a, false, b3, (short)0, acc3, false, false);
    }

    // Store D: VGPR i holds row m0 + i + (lane>=16 ? 8 : 0), col = ncol + 16*tile
    const int rbase = m0 + ((lane >> 4) << 3);
    #pragma unroll
    for (int i = 0; i < 8; ++i) {
        float* __restrict__ hrow = H + (size_t)(rbase + i) * OUT_DIM + ncol;
        hrow[ 0] = acc0[i];
        hrow[16] = acc1[i];
        hrow[32] = acc2[i];
        hrow[48] = acc3[i];
    }
}

// ---------------------------------------------------------------------------
// Kernel 2: COO scatter-add.  agg[r,:] += v * H[c,:].
// One wave (32 lanes) per edge; each lane handles dims (lane, lane+32).
// Edge index forced wave-uniform via readfirstlane so row/col/val become
// s_load (scalar cache) instead of 32 redundant vector loads per wave.
// H (25.6 MB) and agg (25.6 MB) both fit in the 192 MB L2 -> random gathers
// and atomic RMWs stay on-chip; only the 20 MB edge stream hits HBM.
// ---------------------------------------------------------------------------
__global__ void gcn_scatter_atomic(const float* __restrict__ H,
                                   const int*   __restrict__ rows,
                                   const int*   __restrict__ cols,
                                   const float* __restrict__ vals,
                                   float* __restrict__ agg,
                                   int nEdges) {
    const int lane = threadIdx.x & 31;
    const int e    = __builtin_amdgcn_readfirstlane(
                         (blockIdx.x * (blockDim.x >> 5)) + (threadIdx.x >> 5));
    if (e >= nEdges) return;              // uniform guard: EXEC stays all-1s

    const int   r = rows[e];              // uniform address -> s_load_b32
    const int   c = cols[e];
    const float v = vals[e];

    const float* __restrict__ hsrc = H   + (size_t)c * OUT_DIM + lane;
    float*       __restrict__ adst = agg + (size_t)r * OUT_DIM + lane;

    const float m0 = v * hsrc[0];
    const float m1 = v * hsrc[32];

    // relaxed device-scope f32 atomic add -> global_atomic_add_f32 (no return)
    __hip_atomic_fetch_add(adst +  0, m0, __ATOMIC_RELAXED, __HIP_MEMORY_SCOPE_AGENT);
    __hip_atomic_fetch_add(adst + 32, m1, __ATOMIC_RELAXED, __HIP_MEMORY_SCOPE_AGENT);
}

// ---------------------------------------------------------------------------
// Kernel 3: in-place out = relu(out + bias).  OUT_DIM is a power of two.
// ---------------------------------------------------------------------------
__global__ void gcn_bias_relu(float* __restrict__ out,
                              const float* __restrict__ bias,
                              int total) {
    const int i = blockIdx.x * blockDim.x + threadIdx.x;
    if (i >= total) return;
    const float v = out[i] + bias[i & (OUT_DIM - 1)];
    out[i] = v > 0.0f ? v : 0.0f;
}

// ---------------------------------------------------------------------------
// Launcher.  Inputs: X, edge_rows, edge_cols, edge_vals, W, bias.
// Workspace: H (nNodes * OUT_DIM floats).  Accumulator = d_out (zeroed here).
// ---------------------------------------------------------------------------
extern "C" void kernel_launch(void* const* d_in, const int* in_sizes, int n_in,
                              void* d_out, int out_size, void* d_ws, size_t ws_size,
                              hipStream_t stream) {
    const float* X      = (const float*)d_in[0];
    const int*   erow   = (const int*)  d_in[1];
    const int*   ecol   = (const int*)  d_in[2];
    const float* eval   = (const float*)d_in[3];
    const float* W      = (const float*)d_in[4];
    const float* bias   = (const float*)d_in[5];
    float*       out    = (float*)d_out;
    float*       H      = (float*)d_ws;

    const int nNodes = in_sizes[0] / IN_DIM;
    const int nEdges = in_sizes[1];

    // zero the accumulator (d_out) -- graph-capture-safe async memset
    hipMemsetAsync(out, 0, (size_t)out_size * sizeof(float), stream);

    // GEMM: one wave per 16-row strip, 8 waves (256 threads) per block
    const int tilesM       = (nNodes + 15) / 16;
    const int wavesPerBlk  = 256 / 32;
    const int gemmBlocks   = (tilesM + wavesPerBlk - 1) / wavesPerBlk;
    gcn_gemm_wmma<<<gemmBlocks, 256, 0, stream>>>(X, W, H, tilesM);

    // scatter: one wave per edge
    const long long scatterThreads = (long long)nEdges * 32;
    const int scatterBlocks = (int)((scatterThreads + 255) / 256);
    gcn_scatter_atomic<<<scatterBlocks, 256, 0, stream>>>(H, erow, ecol, eval, out, nEdges);

    // bias + relu in place
    const int total = nNodes * OUT_DIM;
    gcn_bias_relu<<<(total + 255) / 256, 256, 0, stream>>>(out, bias, total);
}